// RaftModule_25692494365424
// MI455X (gfx1250) — compile-verified
//
#include <hip/hip_runtime.h>
#include <math.h>

typedef __attribute__((ext_vector_type(2))) float v2f;
typedef __attribute__((ext_vector_type(8))) float v8f;

#define BATCH  2
#define CDEPTH 256
#define HDIM   64
#define WDIM   64
#define PTOT   4096        // HDIM*WDIM
#define TSTEPS 4
#define KWIN   9
#define RWIN   4

// -----------------------------------------------------------------------------
// Kernel 1: all-pairs correlation GEMM via V_WMMA_F32_16X16X4_F32.
//   corr[b, p, q] = (1/16) * sum_c f1[b, c, p] * f2[b, c, q]
//
// Register blocking: one wave computes a 32(p) x 32(q) block = 2x2 grid of
// 16x16 WMMA tiles (4 accumulators). Per K=4 step: 2 A fragments + 2 B
// fragments (8 b32 loads) feed 4 WMMAs -> 2 loads/WMMA (vs 4 unblocked),
// so the L1/TA operand path stays ahead of the f32 matrix pipe.
//
// f1/f2 layout [B][C][P]: lanes stride along P -> coalesced 64B row loads that
// land directly in the 32-bit 16x4 A-operand layout:
//   lane<16  holds M=lane,    K = k+0 / k+1  (VGPR0 / VGPR1)
//   lane>=16 holds M=lane-16, K = k+2 / k+3
// B operand (4x16) mirrors A with N in place of M.
//
// Workgroup = 8 waves (2p x 4q) covering 64(p) x 128(q).
// -----------------------------------------------------------------------------
__global__ __launch_bounds__(256) void corr_gemm_kernel(
    const float* __restrict__ f1, const float* __restrict__ f2,
    float* __restrict__ corr)
{
    const int lane  = threadIdx.x & 31;
    const int wv    = threadIdx.x >> 5;
    const int q0    = blockIdx.x * 128 + (wv & 3) * 32;  // gridDim.x = PTOT/128
    const int p0    = blockIdx.y * 64  + (wv >> 2) * 32; // gridDim.y = PTOT/64
    const int b     = blockIdx.z;

    const float* A = f1 + (size_t)b * CDEPTH * PTOT;
    const float* B = f2 + (size_t)b * CDEPTH * PTOT;

    const int m     = lane & 15;
    const int khalf = lane >> 4;   // 0: K=0/1, 1: K=2/3

    const float* ap = A + (size_t)(2 * khalf) * PTOT + (p0 + m);
    const float* bp = B + (size_t)(2 * khalf) * PTOT + (q0 + m);

    v8f acc00 = {}, acc01 = {}, acc10 = {}, acc11 = {};

    #pragma unroll 2
    for (int k = 0; k < CDEPTH; k += 4) {
        v2f a0, a1, b0, b1;
        a0.x = ap[0];          a0.y = ap[PTOT];        // rows p0..p0+15
        a1.x = ap[16];         a1.y = ap[PTOT + 16];   // rows p0+16..p0+31
        b0.x = bp[0];          b0.y = bp[PTOT];        // cols q0..q0+15
        b1.x = bp[16];         b1.y = bp[PTOT + 16];   // cols q0+16..q0+31

        acc00 = __builtin_amdgcn_wmma_f32_16x16x4_f32(false, a0, false, b0,
                    (short)0, acc00, false, false);
        acc01 = __builtin_amdgcn_wmma_f32_16x16x4_f32(false, a0, false, b1,
                    (short)0, acc01, false, false);
        acc10 = __builtin_amdgcn_wmma_f32_16x16x4_f32(false, a1, false, b0,
                    (short)0, acc10, false, false);
        acc11 = __builtin_amdgcn_wmma_f32_16x16x4_f32(false, a1, false, b1,
                    (short)0, acc11, false, false);

        ap += 4 * PTOT;
        bp += 4 * PTOT;
    }

    // D layout per 16x16 tile: VGPR r -> row (r + 8*khalf), col (lane&15).
    // Scale by 1/sqrt(256) = 1/16.
    float* out = corr + (size_t)b * PTOT * PTOT
                      + (size_t)(p0 + 8 * khalf) * PTOT + (q0 + m);
    #pragma unroll
    for (int r = 0; r < 8; ++r) {
        out[(size_t)r * PTOT]              = acc00[r] * 0.0625f;
        out[(size_t)r * PTOT + 16]         = acc01[r] * 0.0625f;
        out[(size_t)(r + 16) * PTOT]       = acc10[r] * 0.0625f;
        out[(size_t)(r + 16) * PTOT + 16]  = acc11[r] * 0.0625f;
    }
}

// -----------------------------------------------------------------------------
// Kernel 2: windowed bilinear lookup (grid_sample semantics of the reference:
// zero outside [0,63]). One thread per output element [T][B][81][H][W];
// channel kk = ix*9 + iy with dx = kk/9 - 4 applied to x and dy = kk%9 - 4
// applied to y (matching the reference broadcast of d).
// Gathers stay inside the L2-resident 134 MB corr volume.
// -----------------------------------------------------------------------------
__global__ __launch_bounds__(256) void corr_lookup_kernel(
    const float* __restrict__ corr, const float* __restrict__ coords,
    float* __restrict__ out, int total)
{
    int idx = blockIdx.x * blockDim.x + threadIdx.x;
    if (idx >= total) return;

    int hw   = idx & (PTOT - 1);
    int rest = idx >> 12;                 // PTOT = 2^12
    int kk   = rest % (KWIN * KWIN);
    rest    /= (KWIN * KWIN);
    int b    = rest % BATCH;
    int t    = rest / BATCH;

    const float* cslab = coords + (((size_t)t * BATCH + b) * 2) * PTOT;
    float x = cslab[hw];
    float y = cslab[PTOT + hw];

    float px = x + (float)(kk / KWIN - RWIN);
    float py = y + (float)(kk % KWIN - RWIN);

    float x0f = floorf(px);
    float y0f = floorf(py);
    float wx  = px - x0f;
    float wy  = py - y0f;
    int   x0  = (int)x0f;
    int   y0  = (int)y0f;

    const float* cf = corr + ((size_t)b * PTOT + hw) * PTOT;  // 64x64 slice

    auto g = [&](int yi, int xi) -> float {
        if (xi < 0 || xi >= WDIM || yi < 0 || yi >= HDIM) return 0.0f;
        return cf[yi * WDIM + xi];
    };
    float v00 = g(y0,     x0    );
    float v01 = g(y0,     x0 + 1);
    float v10 = g(y0 + 1, x0    );
    float v11 = g(y0 + 1, x0 + 1);

    out[idx] = v00 * (1.0f - wy) * (1.0f - wx)
             + v01 * (1.0f - wy) * wx
             + v10 * wy * (1.0f - wx)
             + v11 * wy * wx;
}

// -----------------------------------------------------------------------------
// Launcher: inputs per setup_inputs(): fmap1 [B,C,H,W] f32, fmap2 [B,C,H,W] f32,
// coords [T,B,2,H,W] f32. Output [T,B,81,H,W] f32.
// d_ws holds the B*P*P fp32 correlation volume (134.2 MB).
// -----------------------------------------------------------------------------
extern "C" void kernel_launch(void* const* d_in, const int* in_sizes, int n_in,
                              void* d_out, int out_size, void* d_ws, size_t ws_size,
                              hipStream_t stream)
{
    const float* f1     = (const float*)d_in[0];
    const float* f2     = (const float*)d_in[1];
    const float* coords = (const float*)d_in[2];
    float*       out    = (float*)d_out;
    float*       corr   = (float*)d_ws;   // needs BATCH*PTOT*PTOT*4 bytes

    dim3 gGrid(PTOT / 128, PTOT / 64, BATCH);   // 32 x 64 x 2 workgroups
    corr_gemm_kernel<<<gGrid, 256, 0, stream>>>(f1, f2, corr);

    int total = TSTEPS * BATCH * (KWIN * KWIN) * PTOT;   // 2,654,208
    corr_lookup_kernel<<<(total + 255) / 256, 256, 0, stream>>>(corr, coords, out, total);
}